// GINE_33578054320563
// MI455X (gfx1250) — compile-verified
//
#include <hip/hip_runtime.h>
#include <hip/hip_bf16.h>

// ---------------------------------------------------------------------------
// GINE forward for gfx1250 (MI455X), fp32 end-to-end using V_WMMA_F32_16X16X4_F32.
// Round 2: hidden-classifier tile staging now uses GLOBAL_LOAD_ASYNC_TO_LDS_B128
// (ASYNCcnt-tracked direct global->LDS DMA) instead of a VGPR round-trip.
// ---------------------------------------------------------------------------

typedef __attribute__((ext_vector_type(2))) float v2f;
typedef __attribute__((ext_vector_type(8))) float v8f;

#define WMMA4(a, b, c) \
    __builtin_amdgcn_wmma_f32_16x16x4_f32(false, (a), false, (b), (short)0, (c), false, false)

__device__ __forceinline__ float leakyf(float x) { return x > 0.f ? x : 0.01f * x; }

static constexpr int N_NODES  = 100000;
static constexpr int N_EDGES  = 1600000;
static constexpr int N_GRAPHS = 2048;
static constexpr int NODE_TILES = N_NODES / 16;   // 6250
static constexpr int EDGE_TILES = N_EDGES / 16;   // 100000

// ---------------------------------------------------------------------------
__global__ void k_zero(float* __restrict__ p, int n) {
    int i = blockIdx.x * blockDim.x + threadIdx.x;
    if (i < n) p[i] = 0.f;
}

// ---------------------------------------------------------------------------
// Fused edge kernel: t = edge_attr[16x16] @ We[16xC];  msg = relu(h[src]+t+be);
// atomic scatter-add into agg[dst]. One wave = 16 edges; WMMA over K=16 (4 steps).
template <int C>
__global__ void k_edge_msg(const float* __restrict__ h,    // [N, C]
                           const float* __restrict__ ea,   // [E, 16]
                           const int*   __restrict__ src,
                           const int*   __restrict__ dst,
                           const float* __restrict__ We,   // [16, C]
                           const float* __restrict__ be,   // [C]
                           float*       __restrict__ agg,  // [N, C]
                           int nTiles) {
    const int lane = threadIdx.x & 31;
    const int wave = threadIdx.x >> 5;
    const int tile = blockIdx.x * (blockDim.x >> 5) + wave;
    if (tile >= nTiles) return;                 // wave-uniform: EXEC stays all-1s
    const int e0    = tile * 16;
    const int mrow  = lane & 15;
    const int khalf = (lane >> 4) * 2;          // 0 or 2

    // Prefetch next tile's edge_attr rows (global_prefetch_b8)
    if (tile + 1 < nTiles)
        __builtin_prefetch(ea + (size_t)(e0 + 16 + mrow) * 16, 0, 1);

    // A fragments: 16 edges x K=16 (4 chained k-steps of K=4)
    v2f afrag[4];
#pragma unroll
    for (int kk = 0; kk < 4; ++kk) {
        const float* ar = ea + (size_t)(e0 + mrow) * 16 + kk * 4 + khalf;
        afrag[kk].x = ar[0];
        afrag[kk].y = ar[1];
    }

#pragma unroll
    for (int nt = 0; nt < C / 16; ++nt) {
        v8f acc = {};
#pragma unroll
        for (int kk = 0; kk < 4; ++kk) {
            const int k0 = kk * 4 + khalf;
            v2f b;
            b.x = We[(size_t)k0 * C + nt * 16 + mrow];
            b.y = We[(size_t)(k0 + 1) * C + nt * 16 + mrow];
            acc = WMMA4(afrag[kk], b, acc);
        }
        const int   col = nt * 16 + mrow;
        const float bev = be[col];
#pragma unroll
        for (int r = 0; r < 8; ++r) {
            const int em = r + ((lane >> 4) << 3);          // edge row 0..15
            const int s  = src[e0 + em];                    // uniform per half-wave
            const int d  = dst[e0 + em];
            float v = acc[r] + bev + h[(size_t)s * C + col]; // coalesced 16-wide gather
            v = v > 0.f ? v : 0.f;                           // relu
            atomicAdd(&agg[(size_t)d * C + col], v);         // L2-resident scatter
        }
    }
}

// ---------------------------------------------------------------------------
// Node update: z = (1+eps)*h + agg;  z @ W[CINx64] + b;  BN -> leaky -> leaky.
// One wave = 16 nodes. z tile staged in LDS (coalesced), WMMA over K=CIN.
template <int CIN, int WAVES>
__global__ void k_node_update(const float* __restrict__ h,    // [N, CIN]
                              const float* __restrict__ agg,  // [N, CIN]
                              const float* __restrict__ epsp, // scalar
                              const float* __restrict__ W,    // [CIN, 64]
                              const float* __restrict__ bias, // [64]
                              const float* __restrict__ g,
                              const float* __restrict__ bb,
                              const float* __restrict__ mM,
                              const float* __restrict__ vV,
                              float*       __restrict__ hout, // [N, 64]
                              int nTiles) {
    __shared__ float zb[WAVES][16 * CIN];
    const int lane = threadIdx.x & 31;
    const int wave = threadIdx.x >> 5;
    const int tile = blockIdx.x * WAVES + wave;
    if (tile >= nTiles) return;
    const int   n0  = tile * 16;
    const float ep1 = 1.0f + epsp[0];

#pragma unroll
    for (int it = 0; it < (16 * CIN) / 32; ++it) {      // coalesced contiguous stage
        int    idx  = it * 32 + lane;
        size_t gidx = (size_t)n0 * CIN + idx;
        zb[wave][idx] = ep1 * h[gidx] + agg[gidx];
    }
    const int mrow  = lane & 15;
    const int khalf = (lane >> 4) * 2;
#pragma unroll
    for (int nt = 0; nt < 4; ++nt) {
        v8f acc = {};
#pragma unroll
        for (int kk = 0; kk < CIN / 4; ++kk) {
            const int k0 = kk * 4 + khalf;
            v2f a, b;
            a.x = zb[wave][mrow * CIN + k0];
            a.y = zb[wave][mrow * CIN + k0 + 1];
            b.x = W[(size_t)k0 * 64 + nt * 16 + mrow];
            b.y = W[(size_t)(k0 + 1) * 64 + nt * 16 + mrow];
            acc = WMMA4(a, b, acc);
        }
        const int   col = nt * 16 + mrow;
        const float inv = __frsqrt_rn(vV[col] + 1e-5f);
        const float gc = g[col], bbc = bb[col], mc = mM[col], bc = bias[col];
#pragma unroll
        for (int r = 0; r < 8; ++r) {
            const int row = n0 + r + ((lane >> 4) << 3);
            float z = acc[r] + bc;
            z = (z - mc) * inv * gc + bbc;   // BatchNorm (eval)
            z = leakyf(z);                    // conv-internal LeakyReLU
            z = leakyf(z);                    // model-level LeakyReLU
            hout[(size_t)row * 64 + col] = z;
        }
    }
}

// ---------------------------------------------------------------------------
__global__ void k_pool(const float* __restrict__ h, const int* __restrict__ batch,
                       float* __restrict__ pool, int n) {   // n = N*64
    int i = blockIdx.x * blockDim.x + threadIdx.x;
    if (i < n) {
        int node = i >> 6, c = i & 63;
        atomicAdd(&pool[(size_t)batch[node] * 64 + c], h[i]);
    }
}

// ---------------------------------------------------------------------------
// Classifier layer 1: h_cat[N,384] @ Wc1[384,128] + bc1 (no activation).
// h_cat rows assembled on the fly from 5 conv outputs + pooled gather.
template <int WAVES>
__global__ void k_cls1(const float* __restrict__ h0, const float* __restrict__ h1,
                       const float* __restrict__ h2, const float* __restrict__ h3,
                       const float* __restrict__ h4, const float* __restrict__ pool,
                       const int*   __restrict__ batch,
                       const float* __restrict__ Wc,   // [384, 128]
                       const float* __restrict__ bc,   // [128]
                       float*       __restrict__ zout, // [N, 128]
                       int nTiles) {
    __shared__ float ab[WAVES][16 * 64];
    const int lane = threadIdx.x & 31;
    const int wave = threadIdx.x >> 5;
    const int tile = blockIdx.x * WAVES + wave;
    if (tile >= nTiles) return;
    const int n0    = tile * 16;
    const int mrow  = lane & 15;
    const int khalf = (lane >> 4) * 2;

    v8f acc[8];
#pragma unroll
    for (int i = 0; i < 8; ++i) { v8f z = {}; acc[i] = z; }

    const float* srcs[6] = {h0, h1, h2, h3, h4, pool};
    for (int blk = 0; blk < 6; ++blk) {
        const float* sp = srcs[blk];
#pragma unroll
        for (int it = 0; it < (16 * 64) / 32; ++it) {
            int    idx  = it * 32 + lane;
            int    row  = idx >> 6, c = idx & 63;
            int    nrow = n0 + row;
            size_t srow = (blk == 5) ? (size_t)batch[nrow] : (size_t)nrow;
            ab[wave][idx] = sp[srow * 64 + c];
        }
#pragma unroll
        for (int nt = 0; nt < 8; ++nt) {
#pragma unroll
            for (int kk = 0; kk < 16; ++kk) {
                const int k0 = kk * 4 + khalf;
                const int kg = blk * 64 + k0;
                v2f a, b;
                a.x = ab[wave][mrow * 64 + k0];
                a.y = ab[wave][mrow * 64 + k0 + 1];
                b.x = Wc[(size_t)kg * 128 + nt * 16 + mrow];
                b.y = Wc[(size_t)(kg + 1) * 128 + nt * 16 + mrow];
                acc[nt] = WMMA4(a, b, acc[nt]);
            }
        }
    }
#pragma unroll
    for (int nt = 0; nt < 8; ++nt) {
        const int   col = nt * 16 + mrow;
        const float bcv = bc[col];
#pragma unroll
        for (int r = 0; r < 8; ++r) {
            const int row = n0 + r + ((lane >> 4) << 3);
            zout[(size_t)row * 128 + col] = acc[nt][r] + bcv;
        }
    }
}

// ---------------------------------------------------------------------------
// Hidden classifier layer: leaky(z[N,128] @ W[128,128] + b).
// A-tile (16x128 fp32, contiguous) staged via GLOBAL_LOAD_ASYNC_TO_LDS_B128:
// direct global->LDS DMA on ASYNCcnt, no VGPR round-trip; one s_wait_asynccnt 0
// before the LDS-resident WMMA A-fragment reads.
template <int WAVES>
__global__ void k_cls_hidden(const float* __restrict__ zin,  // [N, 128]
                             const float* __restrict__ W,    // [128, 128]
                             const float* __restrict__ bc,   // [128]
                             float*       __restrict__ zout, // [N, 128]
                             int nTiles) {
    __shared__ float ab[WAVES][16 * 128];
    const int lane = threadIdx.x & 31;
    const int wave = threadIdx.x >> 5;
    const int tile = blockIdx.x * WAVES + wave;
    if (tile >= nTiles) return;
    const int n0    = tile * 16;
    const int mrow  = lane & 15;
    const int khalf = (lane >> 4) * 2;

    {
        // 16*128 floats = 512 float4 chunks; 32 lanes x 16 steps, one b128 each.
        const float*   gbase   = zin + (size_t)n0 * 128;
        const uint32_t ldsbase = (uint32_t)(uintptr_t)(&ab[wave][0]); // LDS aperture low bits
#pragma unroll
        for (int it = 0; it < 16; ++it) {
            const int      e4    = (it * 32 + lane) * 4;          // float index
            const uint32_t laddr = ldsbase + (uint32_t)e4 * 4u;   // LDS byte address
            const unsigned long long gaddr =
                (unsigned long long)(uintptr_t)(gbase + e4);
            asm volatile("global_load_async_to_lds_b128 %0, %1, off"
                         :: "v"(laddr), "v"(gaddr)
                         : "memory");
        }
        asm volatile("s_wait_asynccnt 0" ::: "memory");
    }

#pragma unroll
    for (int nt = 0; nt < 8; ++nt) {
        v8f acc = {};
#pragma unroll
        for (int kk = 0; kk < 32; ++kk) {
            const int k0 = kk * 4 + khalf;
            v2f a, b;
            a.x = ab[wave][mrow * 128 + k0];
            a.y = ab[wave][mrow * 128 + k0 + 1];
            b.x = W[(size_t)k0 * 128 + nt * 16 + mrow];
            b.y = W[(size_t)(k0 + 1) * 128 + nt * 16 + mrow];
            acc = WMMA4(a, b, acc);
        }
        const int   col = nt * 16 + mrow;
        const float bcv = bc[col];
#pragma unroll
        for (int r = 0; r < 8; ++r) {
            const int row = n0 + r + ((lane >> 4) << 3);
            zout[(size_t)row * 128 + col] = leakyf(acc[r] + bcv);
        }
    }
}

// ---------------------------------------------------------------------------
__global__ void k_final(const float* __restrict__ z, const float* __restrict__ Wf,
                        const float* __restrict__ bf, float* __restrict__ out, int n) {
    int i = blockIdx.x * blockDim.x + threadIdx.x;
    if (i < n) {
        float s = bf[0];
#pragma unroll 8
        for (int k = 0; k < 128; ++k) s += z[(size_t)i * 128 + k] * Wf[k];
        out[i] = 1.f / (1.f + __expf(-s));
    }
}

// ---------------------------------------------------------------------------
extern "C" void kernel_launch(void* const* d_in, const int* in_sizes, int n_in,
                              void* d_out, int out_size, void* d_ws, size_t ws_size,
                              hipStream_t stream) {
    const float* x     = (const float*)d_in[0];
    const int*   eidx  = (const int*)d_in[1];
    const int*   esrc  = eidx;
    const int*   edst  = eidx + N_EDGES;
    const float* ea    = (const float*)d_in[2];
    const int*   batch = (const int*)d_in[3];
    const float* eps1  = (const float*)d_in[4];
    const float* We1   = (const float*)d_in[5];
    const float* be1   = (const float*)d_in[6];
    const float* W1    = (const float*)d_in[7];
    const float* b1    = (const float*)d_in[8];
    const float* g1    = (const float*)d_in[9];
    const float* bb1   = (const float*)d_in[10];
    const float* m1    = (const float*)d_in[11];
    const float* v1    = (const float*)d_in[12];
    const float* epsL  = (const float*)d_in[13];
    const float* WeL   = (const float*)d_in[14];
    const float* beL   = (const float*)d_in[15];
    const float* WL    = (const float*)d_in[16];
    const float* bL    = (const float*)d_in[17];
    const float* gL    = (const float*)d_in[18];
    const float* bbL   = (const float*)d_in[19];
    const float* mL    = (const float*)d_in[20];
    const float* vL    = (const float*)d_in[21];
    const float* Wc1   = (const float*)d_in[22];
    const float* bc1   = (const float*)d_in[23];
    const float* WcL   = (const float*)d_in[24];
    const float* bcL   = (const float*)d_in[25];
    const float* Wf    = (const float*)d_in[26];
    const float* bf    = (const float*)d_in[27];

    const size_t N64 = (size_t)N_NODES * 64;
    float* ws   = (float*)d_ws;
    float* hbuf[5];
    for (int i = 0; i < 5; ++i) hbuf[i] = ws + (size_t)i * N64;
    float* agg  = ws + 5 * N64;
    float* pool = agg + N64;
    float* z1   = pool + (size_t)N_GRAPHS * 64;
    float* z2   = z1 + (size_t)N_NODES * 128;

    auto cdiv = [](int a, int b) { return (a + b - 1) / b; };

    // ---- conv1 (C_in = 32) ----
    k_zero<<<cdiv(N_NODES * 32, 256), 256, 0, stream>>>(agg, N_NODES * 32);
    k_edge_msg<32><<<EDGE_TILES / 8, 256, 0, stream>>>(x, ea, esrc, edst, We1, be1, agg, EDGE_TILES);
    k_node_update<32, 8><<<cdiv(NODE_TILES, 8), 256, 0, stream>>>(
        x, agg, eps1, W1, b1, g1, bb1, m1, v1, hbuf[0], NODE_TILES);

    // ---- stacked convs (C = 64) ----
    for (int i = 0; i < 4; ++i) {
        k_zero<<<cdiv(N_NODES * 64, 256), 256, 0, stream>>>(agg, N_NODES * 64);
        k_edge_msg<64><<<EDGE_TILES / 8, 256, 0, stream>>>(
            hbuf[i], ea, esrc, edst, WeL + (size_t)i * 16 * 64, beL + i * 64, agg, EDGE_TILES);
        k_node_update<64, 8><<<cdiv(NODE_TILES, 8), 256, 0, stream>>>(
            hbuf[i], agg, epsL + i, WL + (size_t)i * 64 * 64, bL + i * 64,
            gL + i * 64, bbL + i * 64, mL + i * 64, vL + i * 64, hbuf[i + 1], NODE_TILES);
    }

    // ---- global_add_pool ----
    k_zero<<<cdiv(N_GRAPHS * 64, 256), 256, 0, stream>>>(pool, N_GRAPHS * 64);
    k_pool<<<cdiv(N_NODES * 64, 256), 256, 0, stream>>>(hbuf[4], batch, pool, N_NODES * 64);

    // ---- classifier ----
    k_cls1<4><<<cdiv(NODE_TILES, 4), 128, 0, stream>>>(
        hbuf[0], hbuf[1], hbuf[2], hbuf[3], hbuf[4], pool, batch, Wc1, bc1, z1, NODE_TILES);
    k_cls_hidden<4><<<cdiv(NODE_TILES, 4), 128, 0, stream>>>(z1, WcL, bcL, z2, NODE_TILES);
    k_cls_hidden<4><<<cdiv(NODE_TILES, 4), 128, 0, stream>>>(
        z2, WcL + 128 * 128, bcL + 128, z1, NODE_TILES);
    k_final<<<cdiv(N_NODES, 256), 256, 0, stream>>>(z1, Wf, bf, (float*)d_out, N_NODES);
}